// DynamicConv_36653250904849
// MI455X (gfx1250) — compile-verified
//
#include <hip/hip_runtime.h>

// ---- problem constants (match reference) ----
#define BB    32
#define C1    128
#define C2    128
#define KEXP  4
#define HH    96
#define WW    96
#define HW    9216        // HH*WW
#define KTOT  1152        // C1*9
#define CHNK  16          // channels per K-chunk in conv
#define KCH   144         // CHNK*9
#define NTILE 96          // spatial columns per block (full image row)
#define NSUB  6           // NTILE/16 accumulators per wave
#define LROW  98          // NTILE + 2 halo
#define WELEM 147456      // C2*C1*9 (per-expert / per-sample weight elems)
#define BNSPLIT 16
#define BNEPS 1e-5f

typedef float v2f __attribute__((ext_vector_type(2)));
typedef float v8f __attribute__((ext_vector_type(8)));

// ---------------- 1) GAP: x[b,c,:,:] mean -> gap[b*C1+c] ----------------
__global__ __launch_bounds__(256) void gap_kernel(const float* __restrict__ x,
                                                  float* __restrict__ gap) {
    __shared__ float red[256];
    const int bc = blockIdx.x;              // 0..B*C1-1
    const float* p = x + (size_t)bc * HW;
    float s = 0.f;
    for (int i = threadIdx.x; i < HW; i += 256) s += p[i];
    red[threadIdx.x] = s;
    __syncthreads();
    for (int st = 128; st > 0; st >>= 1) {
        if (threadIdx.x < st) red[threadIdx.x] += red[threadIdx.x + st];
        __syncthreads();
    }
    if (threadIdx.x == 0) gap[bc] = red[0] * (1.0f / HW);
}

// ---- 2) gating: logits -> softmax -> gate[b,k]; also bmix[b,o] ----
__global__ __launch_bounds__(128) void gate_kernel(const float* __restrict__ gap,
                                                   const float* __restrict__ fc_w,
                                                   const float* __restrict__ fc_b,
                                                   const float* __restrict__ bias,
                                                   float* __restrict__ gate,
                                                   float* __restrict__ bmix) {
    __shared__ float lg[BB * KEXP];
    __shared__ float gt[BB * KEXP];
    const int t = threadIdx.x;              // 128 threads
    {
        const int b = t >> 2, k = t & 3;
        float s = fc_b[k];
        for (int j = 0; j < C1; ++j) s += gap[b * C1 + j] * fc_w[k * C1 + j];
        lg[t] = s;                          // TEMP == 1.0
    }
    __syncthreads();
    if (t < BB) {
        float m = lg[t * 4];
        for (int k = 1; k < 4; ++k) m = fmaxf(m, lg[t * 4 + k]);
        float e[4], sum = 0.f;
        for (int k = 0; k < 4; ++k) { e[k] = __expf(lg[t * 4 + k] - m); sum += e[k]; }
        for (int k = 0; k < 4; ++k) gt[t * 4 + k] = e[k] / sum;
    }
    __syncthreads();
    if (t < BB * KEXP) gate[t] = gt[t];
    for (int i = t; i < BB * C2; i += 128) {
        const int b = i >> 7, o = i & 127;
        float s = 0.f;
        for (int k = 0; k < 4; ++k) s += gt[b * 4 + k] * bias[k * C2 + o];
        bmix[i] = s;
    }
}

// ---- 3) mix expert weights: Wmix[b, idx] = sum_k gate[b,k] * weight[k, idx] ----
__global__ __launch_bounds__(256) void wmix_kernel(const float* __restrict__ gate,
                                                   const float* __restrict__ weight,
                                                   float* __restrict__ Wmix) {
    const int b = blockIdx.y;
    const int idx = blockIdx.x * 256 + threadIdx.x;    // grid.x*256 == WELEM
    const float g0 = gate[b * 4 + 0], g1 = gate[b * 4 + 1];
    const float g2 = gate[b * 4 + 2], g3 = gate[b * 4 + 3];
    const float v = g0 * weight[idx]
                  + g1 * weight[WELEM + idx]
                  + g2 * weight[2 * WELEM + idx]
                  + g3 * weight[3 * WELEM + idx];
    Wmix[(size_t)b * WELEM + idx] = v;
}

// ---- 4) per-sample conv via implicit-GEMM WMMA (f32 16x16x4), + bias ----
// Block: 256 thr = 8 wave32. Grid: (H, B). Wave w -> output channels [16w,16w+16).
// Six 16x16 f32 accumulators per wave cover N = 96 (one full image row),
// so each A fragment (global b64) and tap-offset pair (ds b64) feeds 6 WMMAs.
__global__ __launch_bounds__(256) void conv_wmma_kernel(const float* __restrict__ x,
                                                        const float* __restrict__ Wmix,
                                                        const float* __restrict__ bmix,
                                                        float* __restrict__ y) {
    __shared__ float xs[CHNK * 3 * LROW];    // 4704 floats: input tile + halo
    __shared__ int   offs[KCH];              // k=(c*9+tap) -> LDS offset table

    const int tid  = threadIdx.x;
    const int lane = tid & 31;
    const int wv   = tid >> 5;               // wave 0..7
    const int y0   = blockIdx.x;             // image row
    const int b    = blockIdx.y;             // sample

    if (tid < KCH) {
        const int c = tid / 9, t = tid % 9;
        const int di = t / 3, dj = t % 3;
        offs[tid] = (c * 3 + di) * LROW + dj;
    }

    const int half  = lane >> 4;             // lane half selects K pair
    const int ln    = lane & 15;             // N (and A's M) index
    const int slot0 = half << 1;             // A/B: K = 2*(lane/16) + vgpr
    const int oA    = (wv << 4) + ln;        // A-matrix row (output channel)
    const float* wrowBase = Wmix + ((size_t)(b * C2 + oA)) * KTOT;

    v8f acc[NSUB];
    #pragma unroll
    for (int u = 0; u < NSUB; ++u) acc[u] = (v8f){};

    for (int cc = 0; cc < C1 / CHNK; ++cc) {
        __syncthreads();                     // protects xs reuse + offs build
        const float* xb = x + ((size_t)(b * C1 + cc * CHNK)) * HW;
        for (int idx = tid; idx < CHNK * 3 * LROW; idx += 256) {
            const int c   = idx / (3 * LROW);
            const int rem = idx - c * (3 * LROW);
            const int r   = rem / LROW;
            const int col = rem - r * LROW;
            const int gy  = y0 + r - 1;
            const int gx  = col - 1;
            float v = 0.f;
            if (gy >= 0 && gy < HH && gx >= 0 && gx < WW)
                v = xb[c * HW + gy * WW + gx];
            xs[idx] = v;
        }
        __syncthreads();

        const float* wrow = wrowBase + cc * KCH;
        #pragma unroll 2
        for (int s = 0; s < KCH / 4; ++s) {
            const int kb = (s << 2) + slot0;                 // even -> 8B aligned
            const v2f a = *(const v2f*)(wrow + kb);          // global_load_b64
            const int2 oo = *(const int2*)(&offs[kb]);       // ds_load_b64
            #pragma unroll
            for (int u = 0; u < NSUB; ++u) {
                v2f bf;
                bf.x = xs[oo.x + ln + 16 * u];
                bf.y = xs[oo.y + ln + 16 * u];
                acc[u] = __builtin_amdgcn_wmma_f32_16x16x4_f32(
                    false, a, false, bf, (short)0, acc[u], false, false);
            }
        }
    }

    // D layout: vgpr r, lanes0-15 -> M=r, lanes16-31 -> M=r+8; N = lane%16
    #pragma unroll
    for (int r = 0; r < 8; ++r) {
        const int oo = (wv << 4) + r + (half << 3);
        const float bm = bmix[b * C2 + oo];
        float* yp = y + ((size_t)(b * C2 + oo)) * HW + (size_t)y0 * WW;
        #pragma unroll
        for (int u = 0; u < NSUB; ++u)
            yp[16 * u + ln] = acc[u][r] + bm;
    }
}

// ---- 5a) BN partial sums: grid (C2, BNSPLIT) ----
__global__ __launch_bounds__(256) void bnpartial_kernel(const float* __restrict__ y,
                                                        float* __restrict__ part) {
    __shared__ float rs[256], rs2[256];
    const int c = blockIdx.x, sp = blockIdx.y;
    const int chunk = BB * HW / BNSPLIT;     // 18432
    float s = 0.f, s2 = 0.f;
    for (int i = sp * chunk + threadIdx.x; i < (sp + 1) * chunk; i += 256) {
        const int b = i / HW;
        const int p = i - b * HW;
        const float v = y[((size_t)(b * C2 + c)) * HW + p];
        s += v; s2 += v * v;
    }
    rs[threadIdx.x] = s; rs2[threadIdx.x] = s2;
    __syncthreads();
    for (int st = 128; st > 0; st >>= 1) {
        if (threadIdx.x < st) {
            rs[threadIdx.x]  += rs[threadIdx.x + st];
            rs2[threadIdx.x] += rs2[threadIdx.x + st];
        }
        __syncthreads();
    }
    if (threadIdx.x == 0) {
        part[(c * BNSPLIT + sp) * 2 + 0] = rs[0];
        part[(c * BNSPLIT + sp) * 2 + 1] = rs2[0];
    }
}

// ---- 5b) BN finalize: mean[c], rstd[c] ----
__global__ __launch_bounds__(128) void bnfinal_kernel(const float* __restrict__ part,
                                                      float* __restrict__ mean,
                                                      float* __restrict__ rstd) {
    const int c = threadIdx.x;               // 128 threads == C2
    float s = 0.f, s2 = 0.f;
    for (int sp = 0; sp < BNSPLIT; ++sp) {
        s  += part[(c * BNSPLIT + sp) * 2 + 0];
        s2 += part[(c * BNSPLIT + sp) * 2 + 1];
    }
    const float inv = 1.0f / (float)(BB * HW);
    const float m = s * inv;
    const float var = s2 * inv - m * m;      // ddof=0, matches jnp.var
    mean[c] = m;
    rstd[c] = rsqrtf(var + BNEPS);
}

// ---- 6) BN apply + SiLU, in place on d_out (float4 vectorized) ----
__global__ __launch_bounds__(256) void bnsilu_kernel(float* __restrict__ y,
                                                     const float* __restrict__ mean,
                                                     const float* __restrict__ rstd,
                                                     const float* __restrict__ gamma,
                                                     const float* __restrict__ beta) {
    const size_t i4 = (size_t)blockIdx.x * 256 + threadIdx.x;  // float4 index
    const int c = (int)((i4 / (HW / 4)) & (C2 - 1));
    const float m = mean[c], r = rstd[c], g = gamma[c], bt = beta[c];
    float4 v = ((float4*)y)[i4];
    float* e = (float*)&v;
    #pragma unroll
    for (int j = 0; j < 4; ++j) {
        const float yn = (e[j] - m) * r * g + bt;
        e[j] = yn * (1.0f / (1.0f + __expf(-yn)));             // SiLU
    }
    ((float4*)y)[i4] = v;
}

// ---------------- launch ----------------
extern "C" void kernel_launch(void* const* d_in, const int* in_sizes, int n_in,
                              void* d_out, int out_size, void* d_ws, size_t ws_size,
                              hipStream_t stream) {
    const float* x     = (const float*)d_in[0];
    const float* fc_w  = (const float*)d_in[1];
    const float* fc_b  = (const float*)d_in[2];
    const float* weight= (const float*)d_in[3];
    const float* bias  = (const float*)d_in[4];
    const float* gamma = (const float*)d_in[5];
    const float* beta  = (const float*)d_in[6];
    float* out = (float*)d_out;

    // workspace layout (floats)
    float* ws   = (float*)d_ws;
    float* Wmix = ws;                           // 32*147456 = 4,718,592
    float* gap  = Wmix + (size_t)BB * WELEM;    // 4096
    float* gate = gap  + BB * C1;               // 128
    float* bmix = gate + BB * KEXP;             // 4096
    float* mean = bmix + BB * C2;               // 128
    float* rstd = mean + C2;                    // 128
    float* part = rstd + C2;                    // C2*BNSPLIT*2 = 4096

    gap_kernel<<<BB * C1, 256, 0, stream>>>(x, gap);
    gate_kernel<<<1, 128, 0, stream>>>(gap, fc_w, fc_b, bias, gate, bmix);
    wmix_kernel<<<dim3(WELEM / 256, BB), 256, 0, stream>>>(gate, weight, Wmix);
    conv_wmma_kernel<<<dim3(HH, BB), 256, 0, stream>>>(x, Wmix, bmix, out);
    bnpartial_kernel<<<dim3(C2, BNSPLIT), 256, 0, stream>>>(out, part);
    bnfinal_kernel<<<1, 128, 0, stream>>>(part, mean, rstd);
    bnsilu_kernel<<<(BB * C2 * HW) / 4 / 256, 256, 0, stream>>>(out, mean, rstd, gamma, beta);
}